// DeblurNet_1726576855365
// MI455X (gfx1250) — compile-verified
//
#include <hip/hip_runtime.h>
#include <stdint.h>
#include <stddef.h>

// ---------------------------------------------------------------------------
// Flow-guided 5x5 neighborhood reassembly (DeblurNet warp candidates).
// Phase 2: tiled gather; xi/yi halo tiles staged into LDS via CDNA5 async
//   global->LDS loads (ASYNCcnt), fully-unrolled 25-tap body with CSE'd LDS
//   windows, saddr-form 32-bit-offset NT float4 stores (369MB output stream).
// Phase 1: fused rgb2gray + integer flow-coordinate precompute (streaming).
// ---------------------------------------------------------------------------

namespace {
constexpr int NB = 4;
constexpr int NH = 720;
constexpr int NW = 1280;
constexpr int HW = NH * NW;
constexpr int KP = 25;              // ksize*ksize
constexpr int TW = 64;              // tile width
constexpr int TH = 16;              // tile height
constexpr int LW = TW + 4;          // 68 (halo +-2)
constexpr int LH = TH + 4;          // 20
constexpr int NLD = LW * LH;        // 1360 staged elements per array
}

typedef float v4f __attribute__((ext_vector_type(4)));
typedef int   v4i __attribute__((ext_vector_type(4)));

#if defined(__AMDGCN__) && __has_builtin(__builtin_amdgcn_global_load_async_to_lds_b32)
#define USE_ASYNC_LDS 1
#else
#define USE_ASYNC_LDS 0
#endif

// ---------------------------------------------------------------------------
// Phase 2 compute body (templated on border handling). Fully unrolled so the
// compiler CSEs the overlapping 5x8 LDS windows across the 25 taps. Stores
// use a uniform SGPR base + 32-bit lane offset (25*HW*4 < 2^31).
// ---------------------------------------------------------------------------
template <bool BORDER>
__device__ __forceinline__ void compute_tile(
    const float* __restrict__ grb, const int* s_xi, const int* s_yi,
    v4f pg, float* __restrict__ out_b, int obase, int ty, int lx4, int h, int w0)
{
#pragma unroll
  for (int i = -2; i <= 2; ++i) {
#pragma unroll
    for (int j = -2; j <= 2; ++j) {
      const int p = (i + 2) * 5 + (j + 2);
      const int srow = (ty + 2 - j) * LW + (lx4 + 2 - i);
      v4f res;
#pragma unroll
      for (int c = 0; c < 4; ++c) {
        const int xv = s_xi[srow + c] + i;
        const int yv = s_yi[srow + c] + j;
        const int x0 = xv < 0 ? 0 : (xv > NW - 1 ? NW - 1 : xv);   // v_med3
        const int y0 = yv < 0 ? 0 : (yv > NH - 1 ? NH - 1 : yv);
        // border mask: x0_ < 1 || x0_ > W-2  <=>  (unsigned)(x0_-1) > W-3
        const bool m = ((unsigned)(xv - 1) > (unsigned)(NW - 3)) |
                       ((unsigned)(yv - 1) > (unsigned)(NH - 3));
        const float Ia = grb[y0 * NW + x0];      // near-diagonal -> L2 hits
        float val = m ? pg[c] : Ia;
        if (BORDER) {
          const bool vld = ((unsigned)(h - j) < (unsigned)NH) &
                           ((unsigned)(w0 + c - i) < (unsigned)NW);
          val = vld ? val : 0.0f;
        }
        res[c] = val;
      }
      __builtin_nontemporal_store(res, (v4f*)(out_b + (obase + p * HW)));
    }
  }
}

// ---------------------------------------------------------------------------
// Phase 2: per-block async stage of xi/yi halo tile into LDS, then gather.
// ---------------------------------------------------------------------------
__global__ __launch_bounds__(256) void gather_kernel(
    const float* __restrict__ gray, const float* __restrict__ pgray,
    const int* __restrict__ xi, const int* __restrict__ yi,
    float* __restrict__ out)
{
  __shared__ int s_xi[NLD];
  __shared__ int s_yi[NLD];

  const int b   = blockIdx.z;
  const int tx0 = blockIdx.x * TW;
  const int ty0 = blockIdx.y * TH;
  const int tid = threadIdx.x;

  const int* xib = xi + (size_t)b * HW;
  const int* yib = yi + (size_t)b * HW;

  // Stage (TW+4)x(TH+4) halo tiles of xi/yi with clamped source coordinates.
  for (int e = tid; e < NLD; e += 256) {
    const int ly = e / LW;
    const int lx = e - ly * LW;
    int gy = ty0 + ly - 2; gy = gy < 0 ? 0 : (gy > NH - 1 ? NH - 1 : gy);
    int gx = tx0 + lx - 2; gx = gx < 0 ? 0 : (gx > NW - 1 ? NW - 1 : gx);
    const int gidx = gy * NW + gx;
#if USE_ASYNC_LDS
    __builtin_amdgcn_global_load_async_to_lds_b32(
        (int*)(xib + gidx), &s_xi[e], 0, 0);
    __builtin_amdgcn_global_load_async_to_lds_b32(
        (int*)(yib + gidx), &s_yi[e], 0, 0);
#else
    s_xi[e] = xib[gidx];
    s_yi[e] = yib[gidx];
#endif
  }
#if USE_ASYNC_LDS
#if __has_builtin(__builtin_amdgcn_s_wait_asynccnt)
  __builtin_amdgcn_s_wait_asynccnt(0);
#else
  asm volatile("s_wait_asynccnt 0" ::: "memory");
#endif
#endif
  __syncthreads();

  // Each thread: one row of 4 consecutive pixels (coalesced b128 NT stores).
  const int qx  = tid & 15;
  const int ty  = tid >> 4;            // 0..15
  const int lx4 = qx * 4;
  const int h   = ty0 + ty;
  const int w0  = tx0 + lx4;

  const float* grb  = gray + (size_t)b * HW;
  float*       out_b = out + (size_t)(b * KP) * HW;   // uniform SGPR base
  const int obase = h * NW + w0;                      // 32-bit lane offset

  const v4f pg = *(const v4f*)(pgray + (size_t)b * HW + obase);

  const bool border = (tx0 == 0) | (tx0 == NW - TW) | (ty0 == 0) | (ty0 == NH - TH);
  if (!border) {
    compute_tile<false>(grb, s_xi, s_yi, pg, out_b, obase, ty, lx4, h, w0);
  } else {
    compute_tile<true>(grb, s_xi, s_yi, pg, out_b, obase, ty, lx4, h, w0);
  }
}

// ---------------------------------------------------------------------------
// Phase 1: gray / pgray / trunc(x+flowx) / trunc(y+flowy), 4 pixels per
// thread. 2-D grid (y = batch) so only one cheap const-division remains.
// ---------------------------------------------------------------------------
__global__ __launch_bounds__(256) void prep_kernel(
    const float* __restrict__ img, const float* __restrict__ flow,
    const float* __restrict__ pad,
    float* __restrict__ gray, float* __restrict__ pgray,
    int* __restrict__ xi, int* __restrict__ yi)
{
  const int b    = blockIdx.y;
  const int base = blockIdx.x * 1024 + threadIdx.x * 4;  // < HW, %4 == 0
  const int h    = base / NW;
  const int w    = base - h * NW;

  const size_t ib = ((size_t)(b * 3) * NH + h) * NW + w;   // [B,3,H,W]
  const size_t fb = ((size_t)(b * 2) * NH + h) * NW + w;   // [B,2,H,W]

  const v4f r  = __builtin_nontemporal_load((const v4f*)(img + ib));
  const v4f g  = __builtin_nontemporal_load((const v4f*)(img + ib + (size_t)HW));
  const v4f bl = __builtin_nontemporal_load((const v4f*)(img + ib + (size_t)2 * HW));
  const v4f pr = __builtin_nontemporal_load((const v4f*)(pad + ib));
  const v4f pgc= __builtin_nontemporal_load((const v4f*)(pad + ib + (size_t)HW));
  const v4f pb = __builtin_nontemporal_load((const v4f*)(pad + ib + (size_t)2 * HW));
  const v4f fx = __builtin_nontemporal_load((const v4f*)(flow + fb));
  const v4f fy = __builtin_nontemporal_load((const v4f*)(flow + fb + (size_t)HW));

  const v4f gv  = 0.2989f * r  + 0.587f * g   + 0.114f * bl;
  const v4f pgv = 0.2989f * pr + 0.587f * pgc + 0.114f * pb;

  const v4f wof = {(float)w, (float)(w + 1), (float)(w + 2), (float)(w + 3)};
  const v4f xsf = wof + fx;
  const v4f ysf = (float)h + fy;
  const v4i xv  = __builtin_convertvector(xsf, v4i);   // trunc toward zero
  const v4i yv  = __builtin_convertvector(ysf, v4i);

  // Regular (RT) stores: keep the 59MB of intermediates resident in L2.
  const size_t ob = (size_t)b * HW + base;
  *(v4f*)(gray  + ob) = gv;
  *(v4f*)(pgray + ob) = pgv;
  *(v4i*)(xi    + ob) = xv;
  *(v4i*)(yi    + ob) = yv;
}

// ---------------------------------------------------------------------------
extern "C" void kernel_launch(void* const* d_in, const int* in_sizes, int n_in,
                              void* d_out, int out_size, void* d_ws, size_t ws_size,
                              hipStream_t stream) {
  const float* img  = (const float*)d_in[0];
  const float* flow = (const float*)d_in[1];
  const float* pad  = (const float*)d_in[2];
  // d_in[3] is ksize (==5); kernel is specialized for 5.
  (void)in_sizes; (void)n_in; (void)out_size; (void)ws_size;

  float* gray  = (float*)d_ws;
  float* pgray = gray  + (size_t)NB * HW;
  int*   xi    = (int*)(pgray + (size_t)NB * HW);
  int*   yi    = xi    + (size_t)NB * HW;
  float* out   = (float*)d_out;

  dim3 pgrid(HW / 1024, NB);                      // 900 x 4 blocks
  prep_kernel<<<pgrid, 256, 0, stream>>>(img, flow, pad, gray, pgray, xi, yi);

  dim3 grid(NW / TW, NH / TH, NB);                // 20 x 45 x 4 blocks
  gather_kernel<<<grid, 256, 0, stream>>>(gray, pgray, xi, yi, out);
}